// TopKRouter_40355512714056
// MI455X (gfx1250) — compile-verified
//
#include <hip/hip_runtime.h>
#include <hip/hip_bf16.h>

// ---------------------------------------------------------------------------
// MoE Top-K router for MI455X (gfx1250, wave32).
//   logits = x @ gate_w^T   (16384x2048 * 2048x64)  via V_WMMA_F32_16X16X4_F32
//   probs  = softmax(logits); top-8 + renormalize.
// Bandwidth-bound (x = 134 MB read once -> ~5.8us @ 23.3 TB/s). This version
// double-buffers the K-chunks with CDNA5 async global->LDS loads
// (GLOBAL_LOAD_ASYNC_TO_LDS_B128 / ASYNCcnt) so HBM traffic of chunk k+1
// overlaps the WMMA work on chunk k, with no VGPR staging.
// ---------------------------------------------------------------------------

typedef __attribute__((ext_vector_type(2))) float v2f;
typedef __attribute__((ext_vector_type(4))) float v4f;
typedef __attribute__((ext_vector_type(8))) float v8f;

#define D_MODEL   2048
#define N_EXPERTS 64
#define N_ACTIVE  8
#define ROWS_PER_BLOCK 64      // 4 waves x 16-row WMMA tiles
#define KC        64           // K-chunk staged in LDS
#define NCHUNK    (D_MODEL / KC)
#define LDP       68           // padded LDS pitch (floats): 16B-aligned rows, bank-stride 4
#define THREADS   128

__global__ __launch_bounds__(THREADS)
void topk_router_kernel(const float* __restrict__ x,
                        const float* __restrict__ gw,
                        float* __restrict__ rweights,
                        int*   __restrict__ topk_idx,
                        float* __restrict__ probs_out)
{
    __shared__ float sA[2][ROWS_PER_BLOCK * LDP];   // x tiles (double buffered)
    __shared__ float sB[2][N_EXPERTS * LDP];        // gate_w tiles

    const int tid  = threadIdx.x;
    const int lane = tid & 31;
    const int wave = tid >> 5;
    const int row0 = blockIdx.x * ROWS_PER_BLOCK;

    const float* xblk = x + (size_t)row0 * D_MODEL;

    // Per-thread staging pattern: 8 B128 transfers per tile per chunk.
    //   e4 = i*THREADS + tid ; r = e4>>4 ; c4 = (e4&15)*4   (r is row / expert)
    unsigned gOff[8];                 // global byte offset within chunk-0 tile
    unsigned lOffA[2][8], lOffB[2][8];// per-buffer LDS byte destinations
    {
        unsigned bA0 = (unsigned)(size_t)&sA[0][0];
        unsigned bA1 = (unsigned)(size_t)&sA[1][0];
        unsigned bB0 = (unsigned)(size_t)&sB[0][0];
        unsigned bB1 = (unsigned)(size_t)&sB[1][0];
        #pragma unroll
        for (int i = 0; i < 8; ++i) {
            int e4 = i * THREADS + tid;
            int r  = e4 >> 4;
            int c4 = (e4 & 15) << 2;
            gOff[i]     = (unsigned)((r * D_MODEL + c4) * 4);
            unsigned lo = (unsigned)((r * LDP + c4) * 4);
            lOffA[0][i] = bA0 + lo;
            lOffA[1][i] = bA1 + lo;
            lOffB[0][i] = bB0 + lo;
            lOffB[1][i] = bB1 + lo;
        }
    }

    // Issue async global->LDS B128 loads for one K-chunk into buffer `buf`.
    auto issue_chunk = [&](int chunk, int buf) {
        const unsigned kb = (unsigned)(chunk * KC * 4);
        #pragma unroll
        for (int i = 0; i < 8; ++i) {
            unsigned go = gOff[i] + kb;
            asm volatile("global_load_async_to_lds_b128 %0, %1, %2 offset:0"
                         :: "v"(lOffB[buf][i]), "v"(go), "s"(gw) : "memory");
        }
        #pragma unroll
        for (int i = 0; i < 8; ++i) {
            unsigned go = gOff[i] + kb;
            asm volatile("global_load_async_to_lds_b128 %0, %1, %2 offset:0"
                         :: "v"(lOffA[buf][i]), "v"(go), "s"(xblk) : "memory");
        }
    };

    v8f acc[4] = {v8f{}, v8f{}, v8f{}, v8f{}};   // 16 rows x 64 experts per wave
    const int m  = lane & 15;            // M (A) / N (B) index within 16x16 tile
    const int kh = (lane >> 4) << 1;     // K sub-offset: lanes 0-15 -> K0..1, 16-31 -> K2..3

    // Prologue: fill buffer 0.
    issue_chunk(0, 0);
    asm volatile("s_wait_asynccnt 0x0" ::: "memory");
    __syncthreads();

    for (int k = 0; k < NCHUNK; ++k) {
        const int buf = k & 1;
        if (k + 1 < NCHUNK)
            issue_chunk(k + 1, buf ^ 1);   // overlap HBM with WMMA below

        const float* aRow  = &sA[buf][(wave * 16 + m) * LDP];
        const float* bBase = &sB[buf][0];
        #pragma unroll 4
        for (int ks = 0; ks < KC; ks += 4) {
            v2f a = *(const v2f*)(aRow + ks + kh);
            #pragma unroll
            for (int t = 0; t < 4; ++t) {
                v2f b = *(const v2f*)(bBase + (t * 16 + m) * LDP + ks + kh);
                acc[t] = __builtin_amdgcn_wmma_f32_16x16x4_f32(
                    false, a, false, b, (short)0, acc[t], false, false);
            }
        }
        asm volatile("s_wait_asynccnt 0x0" ::: "memory");
        __syncthreads();
    }

    // --- spill accumulators (logits) into reused sA[0]: row-major [64][LDP] ---
    float* logits = &sA[0][0];
    {
        const int half = lane >> 4;      // C/D layout: lanes 0-15 hold M=v, 16-31 hold M=v+8
        const int n    = lane & 15;
        #pragma unroll
        for (int t = 0; t < 4; ++t)
            #pragma unroll
            for (int v = 0; v < 8; ++v)
                logits[(wave * 16 + v + 8 * half) * LDP + t * 16 + n] = acc[t][v];
    }
    __syncthreads();

    // --- softmax per row (thread-per-row), probs stored back in place ---
    if (tid < ROWS_PER_BLOCK) {
        float* L = logits + tid * LDP;
        float mx = L[0];
        #pragma unroll 8
        for (int e = 1; e < N_EXPERTS; ++e) mx = fmaxf(mx, L[e]);
        float s = 0.f;
        #pragma unroll 8
        for (int e = 0; e < N_EXPERTS; ++e) { float p = __expf(L[e] - mx); L[e] = p; s += p; }
        float inv = 1.0f / s;
        #pragma unroll 8
        for (int e = 0; e < N_EXPERTS; ++e) L[e] *= inv;
    }
    __syncthreads();

    // --- cooperative coalesced store of router_probs (64x64 f32) ---
    #pragma unroll
    for (int i = 0; i < 8; ++i) {
        int e4 = i * THREADS + tid;
        int r  = e4 >> 4;
        int c4 = (e4 & 15) << 2;
        v4f v = *(const v4f*)(logits + r * LDP + c4);
        *(v4f*)(probs_out + (size_t)(row0 + r) * N_EXPERTS + c4) = v;
    }

    // --- top-8 + renormalize (thread-per-row) ---
    if (tid < ROWS_PER_BLOCK) {
        const float* P = logits + tid * LDP;
        unsigned long long sel = 0ull;
        float vals[N_ACTIVE];
        int   idxs[N_ACTIVE];
        float ssum = 0.f;
        #pragma unroll
        for (int k = 0; k < N_ACTIVE; ++k) {
            float best = -1.0f;
            int   bi   = 0;
            for (int e = 0; e < N_EXPERTS; ++e) {
                float p = P[e];
                bool taken = (sel >> e) & 1ull;
                if (!taken && p > best) { best = p; bi = e; }  // strict '>' = lowest-index tie-break
            }
            sel |= 1ull << bi;
            vals[k] = best;
            idxs[k] = bi;
            ssum += best;
        }
        float inv = 1.0f / (ssum + 1e-6f);
        size_t orow = (size_t)(row0 + tid) * N_ACTIVE;
        #pragma unroll
        for (int k = 0; k < N_ACTIVE; ++k) {
            rweights[orow + k] = vals[k] * inv;
            topk_idx[orow + k] = idxs[k];
        }
    }
}

extern "C" void kernel_launch(void* const* d_in, const int* in_sizes, int n_in,
                              void* d_out, int out_size, void* d_ws, size_t ws_size,
                              hipStream_t stream) {
    const float* x  = (const float*)d_in[0];   // (N, 2048) f32
    const float* gw = (const float*)d_in[1];   // (64, 2048) f32
    const int n_rows = in_sizes[0] / D_MODEL;  // 16384

    // d_out layout (flat, return order):
    //   [0,            n*8)   routing_weights  f32
    //   [n*8,          n*16)  topk_idx         i32 (bit pattern)
    //   [n*16, n*16 + n*64)   router_probs     f32
    float* out      = (float*)d_out;
    float* rweights = out;
    int*   topk_idx = (int*)(out + (size_t)n_rows * N_ACTIVE);
    float* probs    = out + (size_t)n_rows * N_ACTIVE * 2;

    dim3 grid(n_rows / ROWS_PER_BLOCK);
    dim3 block(THREADS);
    topk_router_kernel<<<grid, block, 0, stream>>>(x, gw, rweights, topk_idx, probs);
}